// PseudoInverse_19413252178206
// MI455X (gfx1250) — compile-verified
//
#include <hip/hip_runtime.h>
#include <math.h>

// ---------------------------------------------------------------------------
// NUFFT adjoint (KB gridding) + inverse DFT-as-GEMM (WMMA f32) + apod + norm
// Sizes fixed by reference: N=256, K=512, J=6, B=8, M=100000
// Grid is stored INTERLEAVED (re,im) so gridding atomics hit one cacheline
// per tap and pass-1 B-operands load as single b64 per element.
// ---------------------------------------------------------------------------

typedef __attribute__((ext_vector_type(2))) float v2f;
typedef __attribute__((ext_vector_type(8))) float v8f;

#define KK 512
#define NN 256
#define NB 8
#define GRID_ELEMS (KK * KK)          // 262144 complex per batch
#define H_ELEMS (NN * KK)             // 131072 per batch (per plane)
#define TWO_PI 6.28318530717958647692f

__device__ __forceinline__ v8f wmma4(v2f a, v2f b, v8f c) {
  // V_WMMA_F32_16X16X4_F32 : D(16x16) = A(16x4) * B(4x16) + C
  return __builtin_amdgcn_wmma_f32_16x16x4_f32(
      /*neg_a=*/false, a, /*neg_b=*/false, b,
      /*c_mod=*/(short)0, c, /*reuse_a=*/false, /*reuse_b=*/false);
}

// ---- modified Bessel I0 (Abramowitz & Stegun 9.8.1 / 9.8.2, ~1e-7 rel) ----
__device__ __forceinline__ float i0f_dev(float x) {
  float ax = fabsf(x);
  if (ax < 3.75f) {
    float t = (x * x) / (3.75f * 3.75f);
    return 1.0f + t * (3.5156229f + t * (3.0899424f + t * (1.2067492f +
                 t * (0.2659732f + t * (0.0360768f + t * 0.0045813f)))));
  } else {
    float t = 3.75f / ax;
    float p = 0.39894228f + t * (0.01328592f + t * (0.00225319f +
              t * (-0.00157565f + t * (0.00916281f + t * (-0.02057706f +
              t * (0.02635537f + t * (-0.01647633f + t * 0.00392377f)))))));
    return expf(ax) * rsqrtf(ax) * p;
  }
}

__device__ __forceinline__ float kb_beta() {
  // ALPHA = K/N = 2 ; BETA = pi*sqrt((J/ALPHA*(ALPHA-0.5))^2 - 0.8)
  return 3.14159265358979323846f * sqrtf(19.45f);
}

// ---------------------------------------------------------------------------
__global__ void zero_kernel(float* p, int n) {
  int i = blockIdx.x * blockDim.x + threadIdx.x;
  if (i < n) p[i] = 0.0f;
}

// Build DFT matrices with fftshift+crop+apodization+1/512 folded in.
// A1 tables: [i in 0..255][u in 0..511]  (pass 1, transform over grid rows)
// B2 tables: [v in 0..511][j in 0..255]  (pass 2, transform over grid cols)
// Output index i maps to unshifted frequency a(i) = (384+i) mod 512.
// apod d[i] = sinh(sqrt(arg))/sqrt(arg), arg = beta^2 - (pi*J*(i-128)/512)^2
__global__ void build_tables(float* a1c, float* a1s, float* b2c, float* b2s) {
  int t = blockIdx.x * blockDim.x + threadIdx.x;
  if (t >= 2 * H_ELEMS) return;
  const float beta = kb_beta();
  if (t < H_ELEMS) {
    int i = t >> 9;            // output row 0..255
    int u = t & 511;           // grid row 0..511
    int a = (384 + i) & 511;
    int r = (a * u) & 511;     // exact integer angle reduction
    float th = (float)r * (TWO_PI / 512.0f);
    float nf = (float)(i - 128);
    float xf = 3.14159265358979f * 6.0f * nf / 512.0f;
    float arg = beta * beta - xf * xf;          // always > 0 here
    float sq = sqrtf(arg);
    float d = sinhf(sq) / sq;
    float s1 = 1.0f / (512.0f * d);
    a1c[t] = s1 * cosf(th);
    a1s[t] = s1 * sinf(th);
  } else {
    int t2 = t - H_ELEMS;
    int v = t2 >> 8;           // grid col 0..511
    int j = t2 & 255;          // output col 0..255
    int a = (384 + j) & 511;
    int r = (a * v) & 511;
    float th = (float)r * (TWO_PI / 512.0f);
    float nf = (float)(j - 128);
    float xf = 3.14159265358979f * 6.0f * nf / 512.0f;
    float arg = beta * beta - xf * xf;
    float sq = sqrtf(arg);
    float d = sinhf(sq) / sq;
    float s2 = 1.0f / (512.0f * d);
    b2c[t2] = s2 * cosf(th);
    b2s[t2] = -s2 * sinf(th);  // pre-negated: pass 2 only needs -Hi*sin
  }
}

// ---- Kaiser-Bessel gridding: thread = sample m, blockIdx.y = batch b ------
// grid layout: [b][u][v][2] interleaved (re,im)
__global__ void grid_kernel(const float* __restrict__ yr,
                            const float* __restrict__ yi,
                            const float* __restrict__ wt,
                            const float* __restrict__ uv,
                            float* __restrict__ grid, int M) {
  int m = blockIdx.x * blockDim.x + threadIdx.x;
  if (m >= M) return;
  int b = blockIdx.y;
  const float beta = kb_beta();
  const float inv_i0b = 1.0f / i0f_dev(beta);
  const float sc = 512.0f / TWO_PI;

  float ku = uv[2 * m] * sc;
  float kv = uv[2 * m + 1] * sc;
  int ibu = (int)floorf(ku);
  int ibv = (int)floorf(kv);

  float wu[6], wv[6];
  int iu[6], iv[6];
#pragma unroll
  for (int t = 0; t < 6; t++) {
    int ou = ibu + t - 2;
    int ov = ibv + t - 2;
    float du = (float)ou - ku;
    float dv = (float)ov - kv;
    float su = 1.0f - du * du * (1.0f / 9.0f);  // u = 2*dist/J = dist/3
    float sv = 1.0f - dv * dv * (1.0f / 9.0f);
    su = su > 0.0f ? su : 0.0f;
    sv = sv > 0.0f ? sv : 0.0f;
    wu[t] = i0f_dev(beta * sqrtf(su)) * inv_i0b;
    wv[t] = i0f_dev(beta * sqrtf(sv)) * inv_i0b;
    iu[t] = ou & 511;
    iv[t] = ov & 511;
  }

  float w = wt[m];
  float vr = yr[(size_t)b * M + m] * w;
  float vi = yi[(size_t)b * M + m] * w;
  float* gb = grid + (size_t)b * (2 * GRID_ELEMS);
#pragma unroll
  for (int p = 0; p < 6; p++) {
    float wp = wu[p];
    int rowoff = iu[p] << 9;
#pragma unroll
    for (int q = 0; q < 6; q++) {
      float ww = wp * wv[q];
      int lin2 = (rowoff + iv[q]) << 1;
      atomicAdd(gb + lin2, vr * ww);      // re \ same 64B sector
      atomicAdd(gb + lin2 + 1, vi * ww);  // im /
    }
  }
}

// ---- Pass 1: H[i,v] = sum_u A1[i,u] * G[u,v]  (complex), per batch --------
// wave tile: 16(M) x 64(N); 1024 wave tiles = 128 blocks * 8 waves
__global__ void __launch_bounds__(256) pass1_gemm(
    const float* __restrict__ grid,
    const float* __restrict__ a1c, const float* __restrict__ a1s,
    float* __restrict__ hre, float* __restrict__ him) {
  int tid = threadIdx.x;
  int wave = (blockIdx.x << 3) + (tid >> 5);
  int lane = tid & 31;
  int lo = lane & 15;
  int hf = lane >> 4;                 // lane half: K offset 0 or 2
  int b = wave >> 7;
  int rem = wave & 127;
  int mt = rem >> 3;                  // 16 M-tiles
  int nt = rem & 7;                   // 8 N-tiles (64 wide)

  const float* gb = grid + (size_t)b * (2 * GRID_ELEMS);
  int arow = (mt << 4) + lo;
  const float* ac = a1c + arow * 512 + (hf << 1);
  const float* as = a1s + arow * 512 + (hf << 1);
  int ncol = (nt << 6) + lo;

  v8f accR[4] = {};
  v8f accI[4] = {};
  for (int k = 0; k < 512; k += 4) {
    v2f vac = *(const v2f*)(ac + k);   // A: v0=K(k+2hf), v1=K(k+2hf+1)
    v2f vas = *(const v2f*)(as + k);
    int r0 = (k + (hf << 1)) << 10;    // interleaved row stride = 1024 floats
#pragma unroll
    for (int s = 0; s < 4; s++) {
      int c2 = (ncol + (s << 4)) << 1;
      v2f g0 = *(const v2f*)(gb + r0 + c2);          // {Gr,Gi} row k+2hf
      v2f g1 = *(const v2f*)(gb + r0 + 1024 + c2);   // {Gr,Gi} row k+2hf+1
      v2f br = {g0.x, g1.x};
      v2f bi = {g0.y, g1.y};
      v2f bin = -bi;
      accR[s] = wmma4(vac, br, accR[s]);   // Hr += cos*Gr
      accR[s] = wmma4(vas, bin, accR[s]);  // Hr -= sin*Gi
      accI[s] = wmma4(vac, bi, accI[s]);   // Hi += cos*Gi
      accI[s] = wmma4(vas, br, accI[s]);   // Hi += sin*Gr
    }
  }
  float* hrb = hre + (size_t)b * H_ELEMS;
  float* hib = him + (size_t)b * H_ELEMS;
#pragma unroll
  for (int s = 0; s < 4; s++) {
    int c = ncol + (s << 4);
#pragma unroll
    for (int r = 0; r < 8; r++) {
      int row = (mt << 4) + r + (hf << 3);  // C/D: VGPR r = M=r / M=r+8
      hrb[row * 512 + c] = accR[s][r];
      hib[row * 512 + c] = accI[s][r];
    }
  }
}

// ---- Pass 2: out[i,j] = sum_v Hr[i,v]*B2c[v,j] + Hi[i,v]*B2s[v,j] ---------
// (B2s pre-negated). 512 wave tiles = 64 blocks * 8 waves
__global__ void __launch_bounds__(256) pass2_gemm(
    const float* __restrict__ hre, const float* __restrict__ him,
    const float* __restrict__ b2c, const float* __restrict__ b2s,
    float* __restrict__ out) {
  int tid = threadIdx.x;
  int wave = (blockIdx.x << 3) + (tid >> 5);
  int lane = tid & 31;
  int lo = lane & 15;
  int hf = lane >> 4;
  int b = wave >> 6;
  int rem = wave & 63;
  int mt = rem >> 2;                  // 16 M-tiles
  int nt = rem & 3;                   // 4 N-tiles (64 wide)

  const float* hr = hre + (size_t)b * H_ELEMS + ((mt << 4) + lo) * 512 + (hf << 1);
  const float* hi = him + (size_t)b * H_ELEMS + ((mt << 4) + lo) * 512 + (hf << 1);
  int ncol = (nt << 6) + lo;

  v8f acc[4] = {};
  for (int k = 0; k < 512; k += 4) {
    v2f ar = *(const v2f*)(hr + k);
    v2f ai = *(const v2f*)(hi + k);
    int r0 = (k + (hf << 1)) << 8;     // row * 256
#pragma unroll
    for (int s = 0; s < 4; s++) {
      int c = ncol + (s << 4);
      v2f bc = {b2c[r0 + c], b2c[r0 + 256 + c]};
      v2f bs = {b2s[r0 + c], b2s[r0 + 256 + c]};
      acc[s] = wmma4(ar, bc, acc[s]);
      acc[s] = wmma4(ai, bs, acc[s]);
    }
  }
  float* ob = out + (size_t)b * (NN * NN);
#pragma unroll
  for (int s = 0; s < 4; s++) {
    int c = ncol + (s << 4);
#pragma unroll
    for (int r = 0; r < 8; r++) {
      int row = (mt << 4) + r + (hf << 3);
      ob[row * 256 + c] = acc[s][r];
    }
  }
}

// ---- per-batch min/max then normalize -------------------------------------
__global__ void minmax_kernel(const float* __restrict__ x, float* __restrict__ mm) {
  __shared__ float smin[256];
  __shared__ float smax[256];
  int b = blockIdx.x;
  const float* p = x + (size_t)b * (NN * NN);
  float mn = 3.4e38f, mx = -3.4e38f;
  for (int i = threadIdx.x; i < NN * NN; i += 256) {
    float v = p[i];
    mn = fminf(mn, v);
    mx = fmaxf(mx, v);
  }
  smin[threadIdx.x] = mn;
  smax[threadIdx.x] = mx;
  __syncthreads();
  for (int s = 128; s > 0; s >>= 1) {
    if (threadIdx.x < s) {
      smin[threadIdx.x] = fminf(smin[threadIdx.x], smin[threadIdx.x + s]);
      smax[threadIdx.x] = fmaxf(smax[threadIdx.x], smax[threadIdx.x + s]);
    }
    __syncthreads();
  }
  if (threadIdx.x == 0) {
    mm[2 * b] = smin[0];
    mm[2 * b + 1] = smax[0];
  }
}

__global__ void norm_kernel(float* __restrict__ x, const float* __restrict__ mm) {
  int i = blockIdx.x * blockDim.x + threadIdx.x;
  if (i >= NB * NN * NN) return;
  int b = i >> 16;
  float mn = mm[2 * b];
  float mx = mm[2 * b + 1];
  x[i] = (x[i] - mn) / (mx - mn);
}

// ---------------------------------------------------------------------------
extern "C" void kernel_launch(void* const* d_in, const int* in_sizes, int n_in,
                              void* d_out, int out_size, void* d_ws, size_t ws_size,
                              hipStream_t stream) {
  const float* y_real = (const float*)d_in[0];
  const float* y_imag = (const float*)d_in[1];
  const float* weights = (const float*)d_in[2];
  const float* uv = (const float*)d_in[3];
  float* out = (float*)d_out;
  int M = in_sizes[2];  // 100000

  // workspace layout (floats)
  float* w = (float*)d_ws;
  float* grid = w;                                       // 8*512*512*2 interleaved
  float* hre = grid + (size_t)NB * 2 * GRID_ELEMS;       // 8*256*512
  float* him = hre + (size_t)NB * H_ELEMS;
  float* a1c = him + (size_t)NB * H_ELEMS;               // 256*512
  float* a1s = a1c + H_ELEMS;
  float* b2c = a1s + H_ELEMS;                            // 512*256
  float* b2s = b2c + H_ELEMS;
  float* mm = b2s + H_ELEMS;                             // 16

  // 1) zero interleaved grid
  int gz = 2 * NB * GRID_ELEMS;
  zero_kernel<<<(gz + 255) / 256, 256, 0, stream>>>(grid, gz);

  // 2) DFT/apod tables
  build_tables<<<(2 * H_ELEMS) / 256, 256, 0, stream>>>(a1c, a1s, b2c, b2s);

  // 3) KB gridding (atomic scatter; grid lives in L2)
  dim3 gg((M + 255) / 256, NB);
  grid_kernel<<<gg, 256, 0, stream>>>(y_real, y_imag, weights, uv, grid, M);

  // 4) pass 1: complex GEMM over grid rows (WMMA f32 16x16x4)
  pass1_gemm<<<128, 256, 0, stream>>>(grid, a1c, a1s, hre, him);

  // 5) pass 2: real-output GEMM over grid cols (WMMA f32 16x16x4)
  pass2_gemm<<<64, 256, 0, stream>>>(hre, him, b2c, b2s, out);

  // 6) per-batch min/max + normalize
  minmax_kernel<<<NB, 256, 0, stream>>>(out, mm);
  norm_kernel<<<(NB * NN * NN + 255) / 256, 256, 0, stream>>>(out, mm);
}